// CoarsenLatticeModule_25400436588641
// MI455X (gfx1250) — compile-verified
//
#include <hip/hip_runtime.h>

typedef __attribute__((ext_vector_type(2))) float v2f;
typedef __attribute__((ext_vector_type(8))) float v8f;

#define VAL_DIM    64
#define FE         9
#define KDIM       (FE * VAL_DIM)   // 576
#define NR         128
#define M_TILE     64
#define LDS_STRIDE 580              // 576 + 4 pad: rows stay 16B aligned, banks shift by 4/row

__global__ __launch_bounds__(256) void coarsen_wmma_kernel(
    const float* __restrict__ fine,    // [n_fine][64]
    const int*   __restrict__ nbr,     // [n_coarse][9] (int32)
    const float* __restrict__ weight,  // [576][128]
    float*       __restrict__ out,     // [n_coarse][128]
    int n_coarse)
{
    extern __shared__ float Atile[];   // [M_TILE][LDS_STRIDE]

    const int tid = threadIdx.x;
    const int m0  = blockIdx.x * M_TILE;

    // ------- gather phase: async global->LDS, 16 threads x 16B per 256B segment -------
    {
        const int seg_grp = tid >> 4;   // 0..15
        const int l16     = tid & 15;
        for (int s = seg_grp; s < M_TILE * FE; s += 16) {
            const int row  = s / FE;
            const int e    = s - row * FE;
            int grow = m0 + row;
            if (grow >= n_coarse) grow = n_coarse - 1;       // tail safety
            const int idx = nbr[(size_t)grow * FE + e];
            const float* gptr = fine + (size_t)idx * VAL_DIM + l16 * 4;
            // LDS byte offset within our (only) dynamic-LDS allocation
            const unsigned ldsoff =
                (unsigned)((row * LDS_STRIDE + e * VAL_DIM + l16 * 4) * 4);
            asm volatile("global_load_async_to_lds_b128 %0, %1, off"
                         :
                         : "v"(ldsoff), "v"(gptr)
                         : "memory");
        }
        // drain this wave's async ops; barrier below publishes LDS to all waves
        asm volatile("s_wait_asynccnt 0x0" ::: "memory");
    }
    __syncthreads();

    // ------------- WMMA compute: each wave owns a 16-wide N slice -------------
    const int wave  = tid >> 5;        // 0..7
    const int lane  = tid & 31;
    const int h     = lane >> 4;       // lane half
    const int n     = lane & 15;
    const int nbase = wave * 16;

    v8f acc0 = {}; v8f acc1 = {}; v8f acc2 = {}; v8f acc3 = {};

    const float* wcol  = weight + nbase + n;
    const float* abase = Atile + n * LDS_STRIDE;

    #pragma unroll 4
    for (int k0 = 0; k0 < KDIM; k0 += 4) {
        const int ka = k0 + 2 * h;     // K element for VGPR0 (VGPR1 = ka+1)

        v2f b;
        b.x = wcol[(size_t)ka * NR];
        b.y = wcol[(size_t)(ka + 1) * NR];

        v2f a0 = *(const v2f*)(abase + 0 * 16 * LDS_STRIDE + ka);
        v2f a1 = *(const v2f*)(abase + 1 * 16 * LDS_STRIDE + ka);
        v2f a2 = *(const v2f*)(abase + 2 * 16 * LDS_STRIDE + ka);
        v2f a3 = *(const v2f*)(abase + 3 * 16 * LDS_STRIDE + ka);

        acc0 = __builtin_amdgcn_wmma_f32_16x16x4_f32(false, a0, false, b, (short)0, acc0, false, false);
        acc1 = __builtin_amdgcn_wmma_f32_16x16x4_f32(false, a1, false, b, (short)0, acc1, false, false);
        acc2 = __builtin_amdgcn_wmma_f32_16x16x4_f32(false, a2, false, b, (short)0, acc2, false, false);
        acc3 = __builtin_amdgcn_wmma_f32_16x16x4_f32(false, a3, false, b, (short)0, acc3, false, false);
    }

    // ------------- store: C/D layout lane(n)=col, VGPR r + 8h = row -------------
    float* obase = out + (size_t)(m0 + 8 * h) * NR + nbase + n;
    if (m0 + M_TILE <= n_coarse) {
        // full tile: unguarded, non-temporal (output is never re-read; keep L2 for gather)
        #pragma unroll
        for (int r = 0; r < 8; ++r) {
            __builtin_nontemporal_store(acc0[r], obase + (size_t)(0 * 16 + r) * NR);
            __builtin_nontemporal_store(acc1[r], obase + (size_t)(1 * 16 + r) * NR);
            __builtin_nontemporal_store(acc2[r], obase + (size_t)(2 * 16 + r) * NR);
            __builtin_nontemporal_store(acc3[r], obase + (size_t)(3 * 16 + r) * NR);
        }
    } else {
        #pragma unroll
        for (int ms = 0; ms < 4; ++ms) {
            v8f acc = (ms == 0) ? acc0 : (ms == 1) ? acc1 : (ms == 2) ? acc2 : acc3;
            const int rbase = m0 + ms * 16 + 8 * h;
            #pragma unroll
            for (int r = 0; r < 8; ++r) {
                const int orow = rbase + r;
                if (orow < n_coarse)
                    out[(size_t)orow * NR + nbase + n] = acc[r];
            }
        }
    }
}

extern "C" void kernel_launch(void* const* d_in, const int* in_sizes, int n_in,
                              void* d_out, int out_size, void* d_ws, size_t ws_size,
                              hipStream_t stream) {
    const float* fine   = (const float*)d_in[0];
    const int*   nbr    = (const int*)d_in[1];
    const float* weight = (const float*)d_in[2];
    float*       out    = (float*)d_out;

    const int n_coarse = in_sizes[1] / FE;
    const int grid     = (n_coarse + M_TILE - 1) / M_TILE;
    const size_t lds   = (size_t)M_TILE * LDS_STRIDE * sizeof(float);

    hipLaunchKernelGGL(coarsen_wmma_kernel, dim3(grid), dim3(256), lds, stream,
                       fine, nbr, weight, out, n_coarse);
}